// SAModuleMRG_52879637348765
// MI455X (gfx1250) — compile-verified
//
#include <hip/hip_runtime.h>

#define Bc 8
#define Nc 4096
#define Mc 1024
#define Dc 64
#define NTILES_COL (Mc / 16)          // 64 column tiles of lr points
#define NTILE_PAIRS (NTILES_COL / 2)  // 32 pairs (two tiles per b128 load)
#define ROWTILES   (Nc / 16)          // 256 row tiles of hr points per cloud
#define BIGF 3.402823466e+38f

typedef __attribute__((ext_vector_type(2))) float v2f;
typedef __attribute__((ext_vector_type(4))) float v4f;
typedef __attribute__((ext_vector_type(8))) float v8f;

// ---------------------------------------------------------------------------
// Kernel 1: Farthest-point sampling, one workgroup (512 thr, 16 waves) per
// cloud. Each thread owns 8 points IN REGISTERS (hot loop = pure VALU).
// One __syncthreads per step via double-buffered per-wave partials; every
// thread redundantly reduces the 16 broadcast entries (no serial thread-0
// section on the critical path).
// Also packs the WMMA B-operand: tiles pair-interleaved so the NN kernel
// loads one float4 per lane = two 4x16 B tiles:
//   bpack[((b*32+i)*32+lane)*4 + half*2 + {0,1}]
//   lanes 0..15 : (K0=qx, K1=qy)   lanes 16..31 : (K2=qz, K3=|q|^2)
// ---------------------------------------------------------------------------
__global__ __launch_bounds__(512) void fps_kernel(const float* __restrict__ pos,
                                                  int* __restrict__ idx_out,
                                                  float* __restrict__ bpack) {
  __shared__ float sx[Nc], sy[Nc], sz[Nc];   // winner-coordinate lookup only
  __shared__ float rv[2][16];
  __shared__ int   ri[2][16];

  const int b = blockIdx.x;
  const int t = threadIdx.x;
  const int lane = t & 31;
  const int wave = t >> 5;
  const float* pb = pos + (size_t)b * Nc * 3;

  float px[8], py[8], pz[8], mind[8];
#pragma unroll
  for (int k = 0; k < 8; ++k) {
    const int p = t + k * 512;
    const float xx = pb[p * 3 + 0];
    const float yy = pb[p * 3 + 1];
    const float zz = pb[p * 3 + 2];
    px[k] = xx; py[k] = yy; pz[k] = zz;
    sx[p] = xx; sy[p] = yy; sz[p] = zz;
    mind[k] = BIGF;
  }

  if (t == 0) {
    idx_out[b * Mc] = 0;
    const float qx = pb[0], qy = pb[1], qz = pb[2];
    float* tile = bpack + (size_t)(b * NTILE_PAIRS + 0) * 32 * 4;  // i=0, half=0, n=0
    tile[0 * 4 + 0] = qx;
    tile[0 * 4 + 1] = qy;
    tile[16 * 4 + 0] = qz;
    tile[16 * 4 + 1] = qx * qx + qy * qy + qz * qz;
  }
  __syncthreads();

  // last-selected coordinates live in registers (uniform across block)
  float lx = sx[0], ly = sy[0], lz = sz[0];

  for (int s = 1; s < Mc; ++s) {
    const int par = s & 1;
    float bv = -BIGF;
    int   bi = 0;
#pragma unroll
    for (int k = 0; k < 8; ++k) {
      const float dx = px[k] - lx;
      const float dy = py[k] - ly;
      const float dz = pz[k] - lz;
      const float d = dx * dx + dy * dy + dz * dz;
      const float m = fminf(mind[k], d);
      mind[k] = m;
      const bool g = m > bv;                 // strict >: earliest k (lowest p) kept
      bv = g ? m : bv;
      bi = g ? (t + k * 512) : bi;
    }
    // wave32 butterfly argmax, branchless, first-index tie-break
#pragma unroll
    for (int off = 16; off > 0; off >>= 1) {
      const float ov = __shfl_xor(bv, off, 32);
      const int   oi = __shfl_xor(bi, off, 32);
      const bool take = (ov > bv) | ((ov == bv) & (oi < bi));
      bv = take ? ov : bv;
      bi = take ? oi : bi;
    }
    if (lane == 0) { rv[par][wave] = bv; ri[par][wave] = bi; }
    __syncthreads();
    // every thread reduces the 16 broadcast entries (branchless)
    float fbv = rv[par][0];
    int   fbi = ri[par][0];
#pragma unroll
    for (int w = 1; w < 16; ++w) {
      const float ov = rv[par][w];
      const int   oi = ri[par][w];
      const bool take = (ov > fbv) | ((ov == fbv) & (oi < fbi));
      fbv = take ? ov : fbv;
      fbi = take ? oi : fbi;
    }
    lx = sx[fbi]; ly = sy[fbi]; lz = sz[fbi];  // LDS broadcast read
    if (t == 0) {
      idx_out[b * Mc + s] = fbi;
      const int tcol = s >> 4, n = s & 15;
      const int i = tcol >> 1, half = tcol & 1;
      float* tile = bpack + (size_t)(b * NTILE_PAIRS + i) * 32 * 4;
      tile[n * 4 + half * 2 + 0] = lx;
      tile[n * 4 + half * 2 + 1] = ly;
      tile[(16 + n) * 4 + half * 2 + 0] = lz;
      tile[(16 + n) * 4 + half * 2 + 1] = lx * lx + ly * ly + lz * lz;
    }
  }
}

// ---------------------------------------------------------------------------
// Kernel 2: nearest-lr-neighbor search via V_WMMA_F32_16X16X4_F32.
// score[m][n] = |q_n|^2 - 2 p_m . q_n  (same argmin as full d^2; the |p|^2
// term is row-constant and drops out). One wave per 16-row hr tile; the A
// operand stays resident; one global_load_b128 feeds TWO WMMAs.
// ---------------------------------------------------------------------------
__global__ __launch_bounds__(256) void nn_wmma_kernel(const float* __restrict__ pos,
                                                      const float* __restrict__ bpack,
                                                      int* __restrict__ nn) {
  const int lane = threadIdx.x & 31;
  const int w = blockIdx.x * 8 + (threadIdx.x >> 5);   // global row-tile id
  const int b = w >> 8;                                // / ROWTILES
  const int rtile = w & (ROWTILES - 1);
  const int rowBase = rtile * 16;
  const int m = lane & 15;

  // A-matrix 16x4 layout: lanes 0-15 -> (K0,K1), lanes 16-31 -> (K2,K3)
  const int g = (b * Nc + rowBase + m) * 3;
  v2f a;
  if (lane < 16) { a.x = -2.0f * pos[g + 0]; a.y = -2.0f * pos[g + 1]; }
  else           { a.x = -2.0f * pos[g + 2]; a.y = 1.0f; }

  float bestv[8];
  int   besti[8];
#pragma unroll
  for (int r = 0; r < 8; ++r) { bestv[r] = BIGF; besti[r] = 0; }

  const v4f* bptr = (const v4f*)(bpack + (size_t)b * NTILE_PAIRS * 32 * 4);
#pragma unroll 2
  for (int i = 0; i < NTILE_PAIRS; ++i) {
    const v4f bb = bptr[i * 32 + lane];        // two B tiles in one b128 load
    const v2f b0 = { bb.x, bb.y };
    const v2f b1 = { bb.z, bb.w };
    v8f c0 = {}, c1 = {};
    c0 = __builtin_amdgcn_wmma_f32_16x16x4_f32(false, a, false, b0, (short)0, c0, false, false);
    c1 = __builtin_amdgcn_wmma_f32_16x16x4_f32(false, a, false, b1, (short)0, c1, false, false);
    const int col0 = (2 * i) * 16 + m;
    const int col1 = col0 + 16;
#pragma unroll
    for (int r = 0; r < 8; ++r) {
      const bool t0 = c0[r] < bestv[r];        // strict <: earliest tile kept
      bestv[r] = t0 ? c0[r] : bestv[r];
      besti[r] = t0 ? col0 : besti[r];
      const bool t1 = c1[r] < bestv[r];
      bestv[r] = t1 ? c1[r] : bestv[r];
      besti[r] = t1 ? col1 : besti[r];
    }
  }
  // argmin across the 16 lanes sharing each row group (xor<=8 stays in group)
#pragma unroll
  for (int off = 8; off > 0; off >>= 1) {
#pragma unroll
    for (int r = 0; r < 8; ++r) {
      const float ov = __shfl_xor(bestv[r], off, 32);
      const int   oi = __shfl_xor(besti[r], off, 32);
      const bool take = (ov < bestv[r]) | ((ov == bestv[r]) & (oi < besti[r]));
      bestv[r] = take ? ov : bestv[r];
      besti[r] = take ? oi : besti[r];
    }
  }
  if (m == 0) {
    const int rowOff = (lane < 16) ? 0 : 8;    // C/D layout: lanes 16-31 hold M=r+8
#pragma unroll
    for (int r = 0; r < 8; ++r)
      nn[b * Nc + rowBase + rowOff + r] = besti[r];
  }
}

// ---------------------------------------------------------------------------
// Kernel 3: assemble out_x = [x | pos | x_lr[nn] | pos_lr[nn]], zeros, batch.
// ---------------------------------------------------------------------------
__global__ void assemble_kernel(const float* __restrict__ x,
                                const float* __restrict__ pos,
                                const int* __restrict__ idx_ws,
                                const int* __restrict__ nn,
                                float* __restrict__ out) {
  const int totalX = Bc * Nc * 134;
  const int totalZ = Bc * Nc * 3;
  const int total  = totalX + totalZ + Bc * Nc;
  const int i = blockIdx.x * blockDim.x + threadIdx.x;
  if (i >= total) return;

  if (i < totalX) {
    const int row = i / 134;
    const int col = i - row * 134;
    const int b = row >> 12;                  // N = 4096
    float v;
    if (col < 64) {
      v = x[(size_t)row * Dc + col];
    } else if (col < 67) {
      v = pos[(size_t)row * 3 + (col - 64)];
    } else {
      const int nnl  = nn[row];                         // local lr index
      const int orig = b * Nc + idx_ws[b * Mc + nnl];   // original hr point
      if (col < 131) v = x[(size_t)orig * Dc + (col - 67)];
      else           v = pos[(size_t)orig * 3 + (col - 131)];
    }
    out[i] = v;
  } else if (i < totalX + totalZ) {
    out[i] = 0.0f;
  } else {
    const int row = i - totalX - totalZ;
    ((int*)out)[i] = row >> 12;               // batch id = row / N
  }
}

extern "C" void kernel_launch(void* const* d_in, const int* in_sizes, int n_in,
                              void* d_out, int out_size, void* d_ws, size_t ws_size,
                              hipStream_t stream) {
  const float* x   = (const float*)d_in[0];
  const float* pos = (const float*)d_in[1];
  (void)d_in[2]; (void)in_sizes; (void)n_in; (void)out_size; (void)ws_size;

  char* ws = (char*)d_ws;
  int*   idx   = (int*)ws;                                   // B*M ints   (32 KB)
  float* bpack = (float*)(ws + 16384);                       // B*64*64 f  (128 KB)
  int*   nn    = (int*)(ws + 16384 + 131072);                // B*N ints   (128 KB)

  fps_kernel<<<Bc, 512, 0, stream>>>(pos, idx, bpack);
  nn_wmma_kernel<<<(Bc * ROWTILES) / 8, 256, 0, stream>>>(pos, bpack, nn);

  const int total = Bc * Nc * 138;
  assemble_kernel<<<(total + 255) / 256, 256, 0, stream>>>(x, pos, idx, nn, (float*)d_out);
}